// UnPooling_45698452030102
// MI455X (gfx1250) — compile-verified
//
#include <hip/hip_runtime.h>
#include <hip/hip_bf16.h>

// Max-unpool 2x2 (stride 2), NHWC, N=16 H=128 W=128 C=256.
// Pure streaming op: 268MB read (x) + 67MB read (pooled) + 268MB write (out)
// => HBM-bound (~26us at 23.3 TB/s). No matrix math, so no WMMA; optimize the
// data-movement path instead: coalesced b128 NT loads/stores + gfx1250
// global_prefetch_b8 look-ahead.

typedef __attribute__((ext_vector_type(4))) float v4f;

namespace {
constexpr int N  = 16;
constexpr int H  = 128;
constexpr int W  = 128;
constexpr int C  = 256;
constexpr int Hp = H / 2;   // 64
constexpr int Wp = W / 2;   // 64
constexpr int C4 = C / 4;   // 64 float4-groups per channel row
constexpr int BLOCK   = 256;                    // 8 wave32s
constexpr int THREADS = N * Hp * Wp * C4;       // 4,194,304
constexpr int GRID    = THREADS / BLOCK;        // 16384
}  // namespace

__global__ __launch_bounds__(BLOCK) void unpool2x2_max_kernel(
    const float* __restrict__ x,
    const float* __restrict__ pooled,
    float* __restrict__ out)
{
    const unsigned tid = blockIdx.x * BLOCK + threadIdx.x;

    // tid = (((n*Hp + hp)*Wp + wp)*C4 + c4); all dims are powers of two.
    const unsigned c4 = tid & (C4 - 1u);
    const unsigned wp = (tid >> 6)  & (Wp - 1u);
    const unsigned hp = (tid >> 12) & (Hp - 1u);
    const unsigned n  = tid >> 18;

    constexpr unsigned rowStride = (unsigned)W * C;  // 32768 floats
    const unsigned xbase = ((n * H + 2u * hp) * W + 2u * wp) * C + c4 * 4u;
    const unsigned pbase = ((n * Hp + hp) * Wp + wp) * C + c4 * 4u;

    // gfx1250 global_prefetch_b8: pull the next hp's row-pair of x toward L2
    // ahead of the streaming front (read-only, low temporal locality).
    __builtin_prefetch(x + xbase + 2u * rowStride, 0, 0);

    // 2x2 window, (dh,dw) row-major order to match TF/JAX argmax tie-break.
    const v4f a = __builtin_nontemporal_load((const v4f*)(x + xbase));                  // (0,0)
    const v4f b = __builtin_nontemporal_load((const v4f*)(x + xbase + C));              // (0,1)
    const v4f c = __builtin_nontemporal_load((const v4f*)(x + xbase + rowStride));      // (1,0)
    const v4f d = __builtin_nontemporal_load((const v4f*)(x + xbase + rowStride + C));  // (1,1)
    const v4f p = __builtin_nontemporal_load((const v4f*)(pooled + pbase));

    v4f o0, o1, o2, o3;

    // First-max argmax (strictly-greater replaces), scatter max(pooled, 0).
#define UNPOOL_COMP(F)                                            \
    {                                                             \
        float best = a.F; int idx = 0;                            \
        if (b.F > best) { best = b.F; idx = 1; }                  \
        if (c.F > best) { best = c.F; idx = 2; }                  \
        if (d.F > best) { best = d.F; idx = 3; }                  \
        const float v = fmaxf(p.F, 0.0f);                         \
        o0.F = (idx == 0) ? v : 0.0f;                             \
        o1.F = (idx == 1) ? v : 0.0f;                             \
        o2.F = (idx == 2) ? v : 0.0f;                             \
        o3.F = (idx == 3) ? v : 0.0f;                             \
    }

    UNPOOL_COMP(x)
    UNPOOL_COMP(y)
    UNPOOL_COMP(z)
    UNPOOL_COMP(w)
#undef UNPOOL_COMP

    // Dense coalesced writes of the full output; NT — never re-read on device.
    __builtin_nontemporal_store(o0, (v4f*)(out + xbase));
    __builtin_nontemporal_store(o1, (v4f*)(out + xbase + C));
    __builtin_nontemporal_store(o2, (v4f*)(out + xbase + rowStride));
    __builtin_nontemporal_store(o3, (v4f*)(out + xbase + rowStride + C));
}

extern "C" void kernel_launch(void* const* d_in, const int* in_sizes, int n_in,
                              void* d_out, int out_size, void* d_ws, size_t ws_size,
                              hipStream_t stream) {
    (void)in_sizes; (void)n_in; (void)out_size; (void)d_ws; (void)ws_size;
    const float* x      = (const float*)d_in[0];   // [16,128,128,256] f32
    const float* pooled = (const float*)d_in[1];   // [16,64,64,256]   f32
    float*       out    = (float*)d_out;           // [16,128,128,256] f32

    unpool2x2_max_kernel<<<dim3(GRID), dim3(BLOCK), 0, stream>>>(x, pooled, out);
}